// ARAPLoss_56229711839859
// MI455X (gfx1250) — compile-verified
//
#include <hip/hip_runtime.h>
#include <stdint.h>

#define BLK 128          // 4 wave32 per block (allows double-buffered LDS < 64 KB)
#define NWAVE (BLK / 32)
#define FBLK 256         // finish-kernel block
#define GRID_BLOCKS 2048
#define KNB 10           // neighbors per point (reference K)
#define SKP 11           // padded LDS row stride (odd -> no bank conflicts)

// ---- CDNA5 async global->LDS (ASYNCcnt path) ----
__device__ __forceinline__ void async_g2l_b32(uint32_t lds_byte_off, const void* gaddr) {
    // GLOBAL_LOAD_ASYNC_TO_LDS_B32: vdst = per-lane LDS byte address, vaddr = 64-bit global addr
    asm volatile("global_load_async_to_lds_b32 %0, %1, off"
                 :: "v"(lds_byte_off), "v"(gaddr)
                 : "memory");
}
__device__ __forceinline__ void wait_async0() {
    asm volatile("s_wait_asynccnt 0x0" ::: "memory");
}
// Low 32 bits of a generic LDS pointer are the workgroup-relative LDS byte offset
__device__ __forceinline__ uint32_t lds_off_of(const void* p) {
    return (uint32_t)(uintptr_t)p;
}

__global__ __launch_bounds__(BLK) void arap_partial_kernel(
    const float* __restrict__ pc,     // [n,3]
    const float* __restrict__ ninit,  // [n,3]
    const int*   __restrict__ idx,    // [n,K]
    const float* __restrict__ dist,   // [n,K]
    const float* __restrict__ wgt,    // [n,K]
    float*       __restrict__ partial,
    int n, float invNK, float invN3)
{
    // double-buffered tiles: ~40 KB total static LDS
    __shared__ int   s_idx[2][BLK * SKP];
    __shared__ float s_d  [2][BLK * SKP];
    __shared__ float s_w  [2][BLK * SKP];
    __shared__ float s_pc [2][BLK * 3];
    __shared__ float s_ip [2][BLK * 3];
    __shared__ float s_red[NWAVE];

    const int tid  = threadIdx.x;
    const int n10  = n * KNB;            // <= 2e7, fits int32
    const int n3   = n * 3;              // <= 6e6, fits int32
    const int step = (int)gridDim.x * BLK;
    const float invK = 1.0f / (float)KNB;

    // issue one tile's async loads into buffer `buf` (fire-and-forget)
    auto stage = [&](int buf, int base) {
        const int ebase = base * KNB;
        #pragma unroll
        for (int j = 0; j < KNB; ++j) {
            const int e  = j * BLK + tid;          // loop-invariant across tiles -> hoisted
            const int ge = ebase + e;
            if (ge < n10) {
                const int row = e / KNB;
                const int col = e - row * KNB;
                const uint32_t loff = (uint32_t)(row * SKP + col) * 4u;
                async_g2l_b32(lds_off_of(&s_idx[buf][0]) + loff, idx  + ge);
                async_g2l_b32(lds_off_of(&s_d  [buf][0]) + loff, dist + ge);
                async_g2l_b32(lds_off_of(&s_w  [buf][0]) + loff, wgt  + ge);
            }
        }
        const int cbase = base * 3;
        #pragma unroll
        for (int j = 0; j < 3; ++j) {
            const int e  = j * BLK + tid;
            const int ge = cbase + e;
            if (ge < n3) {
                const uint32_t loff = (uint32_t)e * 4u;
                async_g2l_b32(lds_off_of(&s_pc[buf][0]) + loff, pc    + ge);
                async_g2l_b32(lds_off_of(&s_ip[buf][0]) + loff, ninit + ge);
            }
        }
    };

    float acc = 0.0f;
    const int base0 = (int)blockIdx.x * BLK;

    if (base0 < n) {
        // pipeline prologue: stage tile 0
        stage(0, base0);
        wait_async0();
        __syncthreads();

        int buf = 0;
        for (int base = base0; base < n; base += step) {
            // --- overlap: kick off next tile's async DMA before computing this one ---
            const int nxt = base + step;
            if (nxt < n) stage(buf ^ 1, nxt);

            // --- compute current tile (bottleneck: L2-resident random gathers) ---
            const int i = base + tid;
            if (i < n) {
                const float px = s_pc[buf][tid * 3 + 0];
                const float py = s_pc[buf][tid * 3 + 1];
                const float pz = s_pc[buf][tid * 3 + 2];
                float part1 = 0.0f;
                float sx = 0.0f, sy = 0.0f, sz = 0.0f;   // sum of gathered pc
                float tx = 0.0f, ty = 0.0f, tz = 0.0f;   // sum of gathered nn_init
                const int rb = tid * SKP;
                #pragma unroll
                for (int k = 0; k < KNB; ++k) {
                    const uint32_t j2 = (uint32_t)s_idx[buf][rb + k];
                    const float* q = pc + 3u * j2;        // 24 MB table: L2-resident
                    const float qx = q[0], qy = q[1], qz = q[2];
                    const float dx = px - qx, dy = py - qy, dz = pz - qz;
                    const float sq = fmaf(dx, dx, fmaf(dy, dy, dz * dz));
                    part1 += fabsf((sq - s_d[buf][rb + k]) * s_w[buf][rb + k]);
                    sx += qx; sy += qy; sz += qz;
                    const float* r = ninit + 3u * j2;     // 24 MB table: L2-resident
                    tx += r[0]; ty += r[1]; tz += r[2];
                }
                const float ax = (sx - tx) * invK;
                const float ay = (sy - ty) * invK;
                const float az = (sz - tz) * invK;
                const float lda = fabsf(px - (ax + s_ip[buf][tid * 3 + 0]))
                                + fabsf(py - (ay + s_ip[buf][tid * 3 + 1]))
                                + fabsf(pz - (az + s_ip[buf][tid * 3 + 2]));
                acc += part1 * invNK + lda * invN3;
            }

            // next tile's DMA must be complete & visible before it becomes current
            wait_async0();
            __syncthreads();
            buf ^= 1;
        }
    }

    // ---- block reduction, wave32 ----
    #pragma unroll
    for (int off = 16; off > 0; off >>= 1)
        acc += __shfl_down(acc, off, 32);
    const int lane = tid & 31;
    const int wv   = tid >> 5;
    if (lane == 0) s_red[wv] = acc;
    __syncthreads();
    if (wv == 0) {
        float v = (lane < NWAVE) ? s_red[lane] : 0.0f;
        #pragma unroll
        for (int off = NWAVE / 2; off > 0; off >>= 1)
            v += __shfl_down(v, off, 32);
        if (lane == 0) partial[blockIdx.x] = v;
    }
}

__global__ __launch_bounds__(FBLK) void arap_finish_kernel(
    const float* __restrict__ partial, float* __restrict__ out, int m)
{
    __shared__ float s_red[FBLK / 32];
    const int tid = threadIdx.x;
    float acc = 0.0f;
    for (int i = tid; i < m; i += FBLK)   // fixed order -> bit-deterministic
        acc += partial[i];
    #pragma unroll
    for (int off = 16; off > 0; off >>= 1)
        acc += __shfl_down(acc, off, 32);
    const int lane = tid & 31;
    const int wv   = tid >> 5;
    if (lane == 0) s_red[wv] = acc;
    __syncthreads();
    if (wv == 0) {
        float v = (lane < FBLK / 32) ? s_red[lane] : 0.0f;
        #pragma unroll
        for (int off = (FBLK / 32) / 2; off > 0; off >>= 1)
            v += __shfl_down(v, off, 32);
        if (lane == 0) out[0] = v;
    }
}

extern "C" void kernel_launch(void* const* d_in, const int* in_sizes, int n_in,
                              void* d_out, int out_size, void* d_ws, size_t ws_size,
                              hipStream_t stream) {
    (void)n_in; (void)out_size; (void)ws_size;
    const float* pc    = (const float*)d_in[0];   // pc_transformed   [n,3] f32
    const float* ninit = (const float*)d_in[1];   // nn_init_positions[n,3] f32
    const int*   idx   = (const int*)  d_in[2];   // nn_indices       [n,K] i32
    const float* dist  = (const float*)d_in[3];   // nn_distances     [n,K] f32
    const float* wgt   = (const float*)d_in[4];   // neighbor_weights [n,K] f32

    const int n = in_sizes[0] / 3;
    float* partial = (float*)d_ws;                // <= GRID_BLOCKS floats (8 KB)

    // reciprocals computed on host: no v_div/v_rcp sequences in the kernel
    const float invNK = 1.0f / ((double)n * (double)KNB);
    const float invN3 = 1.0f / ((double)n * 3.0);

    long long need = ((long long)n + BLK - 1) / BLK;
    int blocks = (need < (long long)GRID_BLOCKS) ? (int)need : GRID_BLOCKS;

    arap_partial_kernel<<<blocks, BLK, 0, stream>>>(pc, ninit, idx, dist, wgt,
                                                    partial, n, invNK, invN3);
    arap_finish_kernel<<<1, FBLK, 0, stream>>>(partial, (float*)d_out, blocks);
}